// CrossEntropyLoss_80985903334130
// MI455X (gfx1250) — compile-verified
//
#include <hip/hip_runtime.h>

// BCE-over-gathered-matches reduction for MI455X (gfx1250, wave32).
//
// Memory-bound random gather: 16M scalar f32 loads (~2GB of lines worst case
// -> ~90us at 23.3 TB/s HBM). Strategy: 1024 blocks x 256 threads, each thread
// walks matches grid-stride, fully unrolls the 8-batch loop so 16 loads are in
// flight per match, accumulates in f32, block-reduces via LDS tree (ds ops),
// writes deterministic per-block partials to d_ws. A single-block finalize
// kernel reduces the 1024 partials; the last 32->1 lane reduction uses
// V_WMMA_F32_16X16X4_F32 with A=ones (D[m][n] = sum_k B[k][n], layout-
// agnostic: sum of D row 0 == sum of all B elements) + 4 xor-shuffles.
// Everything is deterministic across replays (no float atomics).

typedef __attribute__((ext_vector_type(2))) float v2f;
typedef __attribute__((ext_vector_type(8))) float v8f;

#define LOG_CLAMP (-100.0f)
#define NB 1024
#define NT 256
#define BATCH 8
#define MATROW 2048
#define MATCOL 2048

__global__ __launch_bounds__(NT) void bce_partial_kernel(
    const float* __restrict__ pred, const float* __restrict__ gt,
    const int2* __restrict__ matches, float* __restrict__ partial, int K) {
  const size_t NM = (size_t)MATROW * (size_t)MATCOL;
  float acc = 0.0f;

  const int stride = gridDim.x * blockDim.x;
  for (int k = blockIdx.x * blockDim.x + threadIdx.x; k < K; k += stride) {
    const int2 m = matches[k];                       // (row, col), 8B load
    const size_t idx = (size_t)m.x * MATCOL + (size_t)m.y;
    const float* __restrict__ pp = pred + idx;
    const float* __restrict__ gp = gt + idx;
#pragma unroll
    for (int b = 0; b < BATCH; ++b) {
      const float p = pp[(size_t)b * NM];            // independent loads,
      const float g = gp[(size_t)b * NM];            // all 16 in flight
      const float lp  = fmaxf(logf(p),     LOG_CLAMP);
      const float l1p = fmaxf(log1pf(-p),  LOG_CLAMP);
      acc -= g * lp + (1.0f - g) * l1p;              // bce accumulate
    }
  }

  // Deterministic block reduction (LDS tree).
  __shared__ float sm[NT];
  sm[threadIdx.x] = acc;
  __syncthreads();
  for (int s = NT / 2; s > 0; s >>= 1) {
    if (threadIdx.x < s) sm[threadIdx.x] += sm[threadIdx.x + s];
    __syncthreads();
  }
  if (threadIdx.x == 0) partial[blockIdx.x] = sm[0];
}

__global__ __launch_bounds__(NT) void bce_finalize_kernel(
    const float* __restrict__ partial, float* __restrict__ out, float invK) {
  __shared__ float sm[NT];
  float acc = 0.0f;
  for (int i = threadIdx.x; i < NB; i += NT) acc += partial[i];
  sm[threadIdx.x] = acc;
  __syncthreads();
  // Tree-reduce 256 -> 32.
  for (int s = NT / 2; s >= 32; s >>= 1) {
    if (threadIdx.x < s) sm[threadIdx.x] += sm[threadIdx.x + s];
    __syncthreads();
  }

  // Wave 0 (full 32-lane EXEC, as WMMA requires) finishes the reduction.
  if (threadIdx.x < 32) {
    const float v = sm[threadIdx.x];

    // A = ones(16x4)  => D[m][n] = sum_k B[k][n]  (independent of B layout).
    // Put the 32 lane partials into B's 64 slots (one slot per lane, rest 0):
    // sum over n of D[0][n] == sum of all B elements == total.
    v2f a;  a[0] = 1.0f; a[1] = 1.0f;
    v2f bm; bm[0] = v;   bm[1] = 0.0f;
    v8f c = {};
    c = __builtin_amdgcn_wmma_f32_16x16x4_f32(
        /*neg_a=*/false, a, /*neg_b=*/false, bm,
        /*c_mod=*/(short)0, c, /*reuse_a=*/false, /*reuse_b=*/false);

    // c[0] on lanes 0..15 holds column-sums D[0][n]; sum them across lanes.
    float r = c[0];
    r += __shfl_xor(r, 8, 32);
    r += __shfl_xor(r, 4, 32);
    r += __shfl_xor(r, 2, 32);
    r += __shfl_xor(r, 1, 32);

    if (threadIdx.x == 0) out[0] = r * invK;
  }
}

extern "C" void kernel_launch(void* const* d_in, const int* in_sizes, int n_in,
                              void* d_out, int out_size, void* d_ws, size_t ws_size,
                              hipStream_t stream) {
  const float* pred    = (const float*)d_in[0];   // (8, 2048, 2048) f32
  const float* gt      = (const float*)d_in[1];   // (8, 2048, 2048) f32
  const int2*  matches = (const int2*)d_in[2];    // (K, 2) int32
  const int K = in_sizes[2] / 2;

  float* partial = (float*)d_ws;                  // NB * 4 bytes of scratch
  float* outf    = (float*)d_out;                 // scalar f32

  bce_partial_kernel<<<NB, NT, 0, stream>>>(pred, gt, matches, partial, K);
  bce_finalize_kernel<<<1, NT, 0, stream>>>(partial, outf, 1.0f / (float)K);
}